// TopK_33638183862748
// MI455X (gfx1250) — compile-verified
//
#include <hip/hip_runtime.h>
#include <hip/hip_bf16.h>
#include <cstdint>

#define NUM_EXPERTS 256
#define TOPK_K 8
#define TOPK_GROUP 4
#define ROUTED_SCALE 2.5f
#define TOKENS_PER_BLOCK 8

// One wave32 per token. Lane l owns experts [8l, 8l+8); group g = lane>>2.
__global__ __launch_bounds__(256) void TopK_33638183862748_kernel(
    const float* __restrict__ logits,     // [T, 256]
    const float* __restrict__ bias,       // [256]
    float* __restrict__ out_w,            // [T, 8]
    int*   __restrict__ out_id,           // [T, 8]
    int T)
{
    __shared__ float lds_buf[TOKENS_PER_BLOCK * NUM_EXPERTS];

    const int wave  = threadIdx.x >> 5;
    const int lane  = threadIdx.x & 31;
    const int token = blockIdx.x * TOKENS_PER_BLOCK + wave;
    if (token >= T) return;   // wave-uniform exit

    const float* grow = logits + (size_t)token * NUM_EXPERTS + lane * 8;
    float* lrow = &lds_buf[wave * NUM_EXPERTS + lane * 8];
    // Low 32 bits of a flat LDS pointer == LDS byte address (shared aperture).
    unsigned lds_off = (unsigned)(uintptr_t)(void*)lrow;

    // CDNA5 async copy global->LDS (ASYNCcnt-tracked), 32B per lane = 1KB/row.
    asm volatile(
        "global_load_async_to_lds_b128 %0, %1, off\n\t"
        "global_load_async_to_lds_b128 %0, %1, off offset:16"
        :: "v"(lds_off), "v"(grow) : "memory");

    // Overlap: fetch this lane's 8 bias values while the async copy runs.
    float4 b0 = *(const float4*)(bias + lane * 8);
    float4 b1 = *(const float4*)(bias + lane * 8 + 4);

    asm volatile("s_wait_asynccnt 0" ::: "memory");

    float4 a0 = *(const float4*)(lrow);
    float4 a1 = *(const float4*)(lrow + 4);

    float lg[8] = {a0.x, a0.y, a0.z, a0.w, a1.x, a1.y, a1.z, a1.w};
    float bs[8] = {b0.x, b0.y, b0.z, b0.w, b1.x, b1.y, b1.z, b1.w};
    float sc[8];
#pragma unroll
    for (int j = 0; j < 8; ++j) sc[j] = lg[j] + bs[j];

    const float NEG_INF = -__builtin_inff();

    // ---- lane-local top-2 of biased scores (branchless) ----
    float m1 = NEG_INF, m2 = NEG_INF;
#pragma unroll
    for (int j = 0; j < 8; ++j) {
        float v = sc[j];
        m2 = fmaxf(m2, fminf(m1, v));
        m1 = fmaxf(m1, v);
    }
    // ---- merge top-2 across the 4 lanes of this group ----
#pragma unroll
    for (int d = 1; d <= 2; d <<= 1) {
        float o1 = __shfl_xor(m1, d, 32);
        float o2 = __shfl_xor(m2, d, 32);
        float nm1 = fmaxf(m1, o1);
        float nm2 = fmaxf(fminf(m1, o1), fmaxf(m2, o2));
        m1 = nm1; m2 = nm2;
    }
    float gscore = m1 + m2;           // score of group (lane>>2), same on its 4 lanes

    // ---- top-4 groups: rank with top_k tie rule (lower index wins ties) ----
    const int myg = lane >> 2;
    int rank = 0;
#pragma unroll
    for (int h = 0; h < 8; ++h) {
        float gh = __shfl(gscore, h * 4, 32);
        rank += ((gh > gscore) || (gh == gscore && h < myg)) ? 1 : 0;
    }
    const bool sel = (rank < TOPK_GROUP);
#pragma unroll
    for (int j = 0; j < 8; ++j) sc[j] = sel ? sc[j] : NEG_INF;

    // ---- top-8 experts: 8 rounds of wave argmax over (score, idx, raw logit) ----
    float outw = 0.0f, sum = 0.0f;
    int outi = 0;
#pragma unroll
    for (int k = 0; k < TOPK_K; ++k) {
        float bv = sc[0], bw = lg[0];
        int bidx = lane * 8;
#pragma unroll
        for (int j = 1; j < 8; ++j) {
            bool t = sc[j] > bv;            // strict > keeps earliest index on ties
            bv = t ? sc[j] : bv;
            bw = t ? lg[j] : bw;
            bidx = t ? (lane * 8 + j) : bidx;
        }
#pragma unroll
        for (int d = 1; d < 32; d <<= 1) {
            float ov = __shfl_xor(bv, d, 32);
            int   oi = __shfl_xor(bidx, d, 32);
            float ow = __shfl_xor(bw, d, 32);
            bool take = (ov > bv) || (ov == bv && oi < bidx);
            bv = take ? ov : bv;
            bidx = take ? oi : bidx;
            bw = take ? ow : bw;
        }
        // winner is wave-uniform now
        sum += bw;                           // unbiased logit of winner
        if (lane == k) { outw = bw; outi = bidx; }
        // clear winner slot (constant-index cndmask chain, no scratch)
        const bool own = (bidx >> 3) == lane;
        const int  bj  = bidx & 7;
#pragma unroll
        for (int j = 0; j < 8; ++j) sc[j] = (own && j == bj) ? NEG_INF : sc[j];
    }

    const float invs = ROUTED_SCALE / sum;
    if (lane < TOPK_K) {
        out_w [(size_t)token * TOPK_K + lane] = outw * invs;
        out_id[(size_t)token * TOPK_K + lane] = outi;
    }
}

extern "C" void kernel_launch(void* const* d_in, const int* in_sizes, int n_in,
                              void* d_out, int out_size, void* d_ws, size_t ws_size,
                              hipStream_t stream) {
    const float* logits = (const float*)d_in[0];   // [T, 256] fp32
    const float* bias   = (const float*)d_in[1];   // [256] fp32
    const int T = in_sizes[0] / NUM_EXPERTS;

    float* out_w  = (float*)d_out;                               // first T*8 floats
    int*   out_id = (int*)((float*)d_out + (size_t)T * TOPK_K);  // next  T*8 int32

    const int blocks = (T + TOKENS_PER_BLOCK - 1) / TOKENS_PER_BLOCK;
    TopK_33638183862748_kernel<<<blocks, 256, 0, stream>>>(logits, bias, out_w, out_id, T);
}